// DeformConv_51436528337589
// MI455X (gfx1250) — compile-verified
//
#include <hip/hip_runtime.h>

// ---------------------------------------------------------------------------
// Deformable conv (B=8, C=128, H=W=64, Cout=128, K=3, pad=1, stride=1), fp32.
// Stage 1: offsets = conv3x3(x, w_off)+b_off  -> d_ws   (8,18,64,64)
// Stage 2: bilinear gather -> implicit GEMM (M=pixels, N=128, K=1152)
//          via v_wmma_f32_16x16x4_f32, with the w_def B-panel DMA'd into LDS
//          by the Tensor Data Mover (tensor_load_to_lds / s_wait_tensorcnt),
//          overlapped with the VALU bilinear gather.
// ---------------------------------------------------------------------------

typedef __attribute__((ext_vector_type(2))) float v2f;
typedef __attribute__((ext_vector_type(8))) float v8f;

constexpr int Cc   = 128;
constexpr int Hh   = 64;
constexpr int Wd   = 64;
constexpr int HW   = Hh * Wd;        // 4096
constexpr int OffC = 18;
constexpr int Kfull = Cc * 9;        // 1152

// ============================ Stage 1 ======================================
__global__ __launch_bounds__(256) void offsets_conv_kernel(
    const float* __restrict__ x, const float* __restrict__ w_off,
    const float* __restrict__ b_off, float* __restrict__ offs)
{
    __shared__ float lw[64 * 9 * OffC];   // [(cLocal*9+k)][o]   41,472 B
    __shared__ float lb[OffC];

    const int tid = threadIdx.x;
    if (tid < OffC) lb[tid] = b_off[tid];
    __syncthreads();

    const int bimg = blockIdx.x >> 4;             // 16 blocks per image
    const int p    = ((blockIdx.x & 15) << 8) | tid;
    const int h    = p >> 6;
    const int w    = p & 63;

    float acc[OffC];
#pragma unroll
    for (int o = 0; o < OffC; ++o) acc[o] = lb[o];

    const float* xb = x + (size_t)bimg * Cc * HW;

    for (int c0 = 0; c0 < Cc; c0 += 64) {
        __syncthreads();
        for (int i = tid; i < OffC * 64 * 9; i += 256) {
            int o = i / 576;           // 576 = 64*9
            int r = i % 576;
            lw[r * OffC + o] = w_off[(size_t)o * Kfull + c0 * 9 + r];
        }
        __syncthreads();

        for (int cl = 0; cl < 64; ++cl) {
            const float* xc = xb + (size_t)(c0 + cl) * HW;
#pragma unroll
            for (int k = 0; k < 9; ++k) {
                int yy = h + (k / 3) - 1;
                int xx = w + (k % 3) - 1;
                float xv = 0.0f;
                if (yy >= 0 && yy < Hh && xx >= 0 && xx < Wd)
                    xv = xc[yy * Wd + xx];
                const float* wr = &lw[(cl * 9 + k) * OffC];
#pragma unroll
                for (int o = 0; o < OffC; ++o)
                    acc[o] = fmaf(xv, wr[o], acc[o]);
            }
        }
    }

    float* ob = offs + (size_t)bimg * OffC * HW + p;
#pragma unroll
    for (int o = 0; o < OffC; ++o) ob[(size_t)o * HW] = acc[o];
}

// ============================ Stage 2 ======================================
// One workgroup (256 threads = 8 waves) = one 16-pixel M-tile x all 128 Cout.
// K = 1152 in 16 chunks of 72 (= 8 channels * 9 taps).
constexpr int CCH = 8;             // channels per chunk
constexpr int KCH = CCH * 9;       // 72 dwords per B row
// TDM pads 1 dword after every 8 dwords -> effective LDS row stride 81,
// element kk lives at offset kk + kk/8 within the row.
constexpr int BROW = KCH + KCH / 8;   // 81
__device__ __forceinline__ int bcol(int kk) { return kk + (kk >> 3); }

#if __has_builtin(__builtin_amdgcn_tensor_load_to_lds)
#define USE_TDM 1
typedef unsigned int u32x4 __attribute__((ext_vector_type(4)));
typedef int          i32x4 __attribute__((ext_vector_type(4)));
typedef int          i32x8 __attribute__((ext_vector_type(8)));
typedef __attribute__((address_space(3))) float lds_f;
#else
#define USE_TDM 0
#endif

__global__ __launch_bounds__(256) void deform_gemm_kernel(
    const float* __restrict__ x, const float* __restrict__ w_def,
    const float* __restrict__ offs, float* __restrict__ out)
{
    __shared__ float sB[Cc * BROW];     // [o][padded kk]  41,472 B
    __shared__ float sA[KCH * 16];      // [kk][pixel]      4,608 B
    __shared__ float sO[Cc * 17];       // D staging        8,704 B
    __shared__ float sWt[4][144];       // tap weights      2,304 B
    __shared__ int   sIx[4][144];       // tap indices      2,304 B

    const int tid  = threadIdx.x;
    const int lane = tid & 31;
    const int wave = tid >> 5;          // 0..7
    const int hf   = (lane >> 4) & 1;   // K-half within wave
    const int l15  = lane & 15;

    const int tile = blockIdx.x;        // 2048 tiles
    const int bimg = tile >> 8;         // 256 tiles per image
    const int seg  = tile & 255;
    const int h    = seg >> 2;          // 4 segments per 64-wide row
    const int w0   = (seg & 3) << 4;

    // ---- per-tile bilinear tap tables: 16 pixels x 9 taps -----------------
    if (tid < 144) {
        const int k  = tid >> 4;        // 0..8
        const int pi = tid & 15;
        const int w  = w0 + pi;
        const size_t obase = (size_t)bimg * OffC * HW + (size_t)h * Wd + w;
        float dy = offs[obase + (size_t)(2 * k)     * HW];
        float dx = offs[obase + (size_t)(2 * k + 1) * HW];
        float py = (float)(h - 1 + k / 3) + dy;
        float px = (float)(w - 1 + k % 3) + dx;
        float y0f = floorf(py), x0f = floorf(px);
        float wy = py - y0f, wx = px - x0f;
        int y0 = (int)y0f, x0 = (int)x0f;
#pragma unroll
        for (int t = 0; t < 4; ++t) {
            int yi = y0 + (t >> 1);
            int xi = x0 + (t & 1);
            float wt = ((t >> 1) ? wy : 1.0f - wy) *
                       ((t & 1)  ? wx : 1.0f - wx);
            bool valid = (yi >= 0) && (yi < Hh) && (xi >= 0) && (xi < Wd);
            int yc = min(max(yi, 0), Hh - 1);
            int xc = min(max(xi, 0), Wd - 1);
            sWt[t][k * 16 + pi] = valid ? wt : 0.0f;
            sIx[t][k * 16 + pi] = yc * Wd + xc;
        }
    }

#if USE_TDM
    // Wave-uniform pieces of the Tensor DMA descriptor (D#).
    const unsigned ldsB = (unsigned)(unsigned long long)(lds_f*)sB;
    // group1: data_size=4B(2), pad_enable=1, pad_interval=8dw(2), pad_amount=1dw(0)
    const i32x8 g1 = { (int)((2u << 16) | (1u << 20) | (2u << 22)),
                       (int)((unsigned)KCH << 16),   // tensor_dim0 = 72
                       (int)((unsigned)Cc  << 16),   // tensor_dim1 = 128
                       (int)((unsigned)KCH << 16),   // tile_dim0 = 72
                       (int)Cc,                      // tile_dim1 = 128 (dim2=0)
                       (int)Kfull,                   // tensor_dim0_stride = 1152
                       0, 0 };
    const i32x4 gz = { 0, 0, 0, 0 };
#endif

    v8f acc = {};
    const float* xb = x + (size_t)bimg * Cc * HW;
    const int n0 = wave * 16;

    for (int c0 = 0; c0 < Cc; c0 += CCH) {
        __syncthreads();   // previous chunk fully consumed / taps ready

#if USE_TDM
        // ---- async DMA of the w_def chunk into sB (wave 0 only) -----------
        if (wave == 0) {
            unsigned long long ga =
                (unsigned long long)(const void*)(w_def + (size_t)c0 * 9);
            u32x4 g0 = { 1u,                       // count=1
                         ldsB,                     // LDS byte address
                         (unsigned)ga,
                         (unsigned)((ga >> 32) & 0x01FFFFFFu) | 0x80000000u };
#if __has_include(<hip/amd_detail/amd_gfx1250_TDM.h>)
            const i32x8 gz8 = { 0, 0, 0, 0, 0, 0, 0, 0 };
            __builtin_amdgcn_tensor_load_to_lds(g0, g1, gz, gz, gz8, 0);
#else
            __builtin_amdgcn_tensor_load_to_lds(g0, g1, gz, gz, 0);
#endif
        }
#else
        // ---- fallback: synchronous copy into the same padded layout -------
        for (int i = tid; i < Cc * KCH; i += 256) {
            int o  = i / KCH;
            int kk = i % KCH;
            sB[o * BROW + bcol(kk)] = w_def[(size_t)o * Kfull + c0 * 9 + kk];
        }
#endif

        // ---- gather A panel: sA[kk][pixel] (overlaps the TDM transfer) ----
        for (int e = tid; e < KCH * 16; e += 256) {
            int pi = e & 15;
            int kk = e >> 4;            // 0..71
            int c  = c0 + kk / 9;
            int k  = kk % 9;
            int ti = k * 16 + pi;
            const float* plane = xb + (size_t)c * HW;
            float v = sWt[0][ti] * plane[sIx[0][ti]];
            v = fmaf(sWt[1][ti], plane[sIx[1][ti]], v);
            v = fmaf(sWt[2][ti], plane[sIx[2][ti]], v);
            v = fmaf(sWt[3][ti], plane[sIx[3][ti]], v);
            sA[kk * 16 + pi] = v;
        }

#if USE_TDM
        if (wave == 0) __builtin_amdgcn_s_wait_tensorcnt(0);
#endif
        __syncthreads();

        // ---- WMMA over K-chunk: 18 x v_wmma_f32_16x16x4_f32 ---------------
#pragma unroll
        for (int kk0 = 0; kk0 < KCH; kk0 += 4) {
            const int ka = kk0 + 2 * hf;
            v2f afr, bfr;
            afr.x = sA[ka * 16 + l15];
            afr.y = sA[(ka + 1) * 16 + l15];
            bfr.x = sB[(n0 + l15) * BROW + bcol(ka)];
            bfr.y = sB[(n0 + l15) * BROW + bcol(ka) + 1];
            acc = __builtin_amdgcn_wmma_f32_16x16x4_f32(
                false, afr, false, bfr, (short)0, acc, false, false);
        }
    }

    // ---- epilogue: transpose D through LDS, coalesced stores --------------
    __syncthreads();
#pragma unroll
    for (int j = 0; j < 8; ++j) {
        int m = j + 8 * hf;             // pixel within tile
        int o = n0 + l15;               // output channel
        sO[o * 17 + m] = acc[j];
    }
    __syncthreads();

    float* ob = out + (size_t)bimg * Cc * HW + (size_t)h * Wd + w0;
    for (int i = tid; i < Cc * 16; i += 256) {
        int o = i >> 4;
        int m = i & 15;
        ob[(size_t)o * HW + m] = sO[o * 17 + m];
    }
}

// ============================ Host side ====================================
extern "C" void kernel_launch(void* const* d_in, const int* in_sizes, int n_in,
                              void* d_out, int out_size, void* d_ws, size_t ws_size,
                              hipStream_t stream) {
    (void)in_sizes; (void)n_in; (void)out_size; (void)ws_size;
    const float* x     = (const float*)d_in[0];
    const float* w_off = (const float*)d_in[1];
    const float* b_off = (const float*)d_in[2];
    const float* w_def = (const float*)d_in[3];
    float* out  = (float*)d_out;
    float* offs = (float*)d_ws;   // 8*18*4096 floats = 2.36 MB

    offsets_conv_kernel<<<dim3(8 * 16), dim3(256), 0, stream>>>(x, w_off, b_off, offs);
    deform_gemm_kernel<<<dim3(2048), dim3(256), 0, stream>>>(x, w_def, offs, out);
}